// ECELoss_66898410602955
// MI455X (gfx1250) — compile-verified
//
#include <hip/hip_runtime.h>
#include <hip/hip_bf16.h>

// ---------------- CDNA5 vector types ----------------
typedef __attribute__((ext_vector_type(2))) float    v2f;
typedef __attribute__((ext_vector_type(8))) float    v8f;
typedef __attribute__((ext_vector_type(4))) unsigned v4u;
typedef __attribute__((ext_vector_type(8))) int      v8i;
typedef __attribute__((ext_vector_type(4))) int      v4i;

// ---------------- compile-time problem constants ----------------
#define NB   15        // number of ECE / adaptive bins
#define NT   14        // number of quantile thresholds (NB-1)
#define HB   65536     // 16-bit histogram bins
#define CHUNK 256      // columns staged per LDS chunk
#define LSTR  260      // LDS row stride in floats (260 % 64 == 4 -> conflict-free b64 reads)
#define TIE_CAP 2048

#if __has_builtin(__builtin_amdgcn_tensor_load_to_lds) && __has_builtin(__builtin_amdgcn_s_wait_tensorcnt)
#define HAVE_TDM 1
#else
#define HAVE_TDM 0
#endif

// ---------------- device-global scratch (avoids ws_size dependence) ----------------
__device__ float    g_logpt[131072];
__device__ unsigned g_key[131072];        // float bits of prob (monotone, prob >= 0)
__device__ int      g_hist1[HB];
__device__ int      g_hist2[NT * HB];
__device__ unsigned g_selb[NT];           // selected high-16 bucket per threshold
__device__ int      g_r1[NT];             // residual rank within high bucket
__device__ unsigned g_tkey[NT];           // full 32-bit key at threshold rank
__device__ int      g_r2[NT];             // residual rank within exact-key group
__device__ unsigned g_tidx[NT];           // original index tie-break at threshold

// ---------------- zero/init kernel ----------------
__global__ void k_zero(float* out) {
    int i = blockIdx.x * blockDim.x + threadIdx.x;
    if (i == 0) *out = 0.0f;
    if (i < HB) g_hist1[i] = 0;
    for (int j = i; j < NT * HB; j += gridDim.x * blockDim.x) g_hist2[j] = 0;
}

#if HAVE_TDM
// ---------------- TDM descriptor: 16 x CHUNK f32 tile, LDS rows padded to LSTR ----------------
// pad_interval = 256 DWORDs (code 7), pad_amount = 4 DWORDs (code 3) -> LDS row stride 260 floats.
__device__ __forceinline__ void tdm_issue(const float* gbase, unsigned lds_byte, int C) {
    const unsigned long long ga = (unsigned long long)(size_t)gbase;
    v4u g0;
    g0[0] = 1u;                                              // count=1, user mode
    g0[1] = lds_byte;                                        // lds_addr (bytes)
    g0[2] = (unsigned)(ga & 0xFFFFFFFFu);                    // global_addr[31:0]
    g0[3] = (unsigned)((ga >> 32) & 0x1FFFFFFu) | (2u << 30);// global_addr[56:32] | type=2
    v8i g1;
    g1[0] = (int)((2u << 16) | (1u << 20) | (7u << 22) | (3u << 25)); // 4B elems, pad enable
    g1[1] = (int)(((unsigned)CHUNK & 0xFFFFu) << 16);        // tensor_dim0[15:0]
    g1[2] = (int)((((unsigned)CHUNK >> 16) & 0xFFFFu) | (16u << 16)); // td0 hi | tensor_dim1=16
    g1[3] = (int)(((unsigned)CHUNK) << 16);                  // td1 hi=0 | tile_dim0=CHUNK
    g1[4] = 16;                                              // tile_dim1=16, tile_dim2=0
    g1[5] = C;                                               // tensor_dim0_stride[31:0] (elems)
    g1[6] = 0;                                               // stride hi | dim1_stride lo
    g1[7] = 0;
    const v4i z4 = {0, 0, 0, 0};
    const v8i z8 = {0, 0, 0, 0, 0, 0, 0, 0};
    // 6-arg form (upstream clang-23 / therock-10.0 headers)
    __builtin_amdgcn_tensor_load_to_lds(g0, g1, z4, z4, z8, 0);
}
#endif

// ---------------- streaming logsumexp: 16 rows / block, TDM staging, WMMA row-sum ----------------
__global__ __launch_bounds__(256) void k_logsumexp(const float* __restrict__ in,
                                                   const int* __restrict__ tgt,
                                                   int N, int C) {
    __shared__ float tile[2][16 * LSTR];
    __shared__ float s_rowmax[16];
    __shared__ float s_scale[16];
    __shared__ float s_rowsum[16];

    const int tid  = threadIdx.x;
    const int lane = tid & 31;
    const int wv   = tid >> 5;                       // 8 waves
    const int wvu  = __builtin_amdgcn_readfirstlane(wv);   // scalar wave id -> uniform loops
    const int row0 = blockIdx.x * 16;
    const int nchunk = (C + CHUNK - 1) / CHUNK;

    if (tid < 16) { s_rowmax[tid] = -__builtin_inff(); s_rowsum[tid] = 0.0f; }

    // warm L2 for the irregular tail chunk (handled by float4 staging, not TDM)
    {
        const int tb = (nchunk - 1) * CHUNK;
        if (C - tb < CHUNK && tid < 128) {
            const int r = tid >> 3, l = tid & 7;
            if (tb + l * 32 < C)
                __builtin_prefetch(in + (size_t)(row0 + r) * C + tb + l * 32, 0, 3);
        }
    }

#if HAVE_TDM
    if (wvu == 0 && C >= CHUNK)                      // chunk 0 is full -> TDM
        tdm_issue(in + (size_t)row0 * C, (unsigned)(size_t)&tile[0][0], C);
#endif

    v8f acc = {};                                    // WMMA C/D: 16x16 f32, rows replicated over N
    const int m16  = lane & 15;                      // A-layout row owned by this lane
    const int kb   = (lane >> 4) << 1;               // A-layout K offset (0 or 2)
    const int half = (lane >> 4) << 3;               // row group for rescale / extraction

    for (int ci = 0; ci < nchunk; ++ci) {
        const int cbase = ci * CHUNK;
        const int cc    = min(CHUNK, C - cbase);
        const bool fullc = (cc == CHUNK);
        const int  nx    = ci + 1;
        const bool fulln = (nx < nchunk) && (C - nx * CHUNK >= CHUNK);
        float* buf = tile[ci & 1];

#if HAVE_TDM
        if (wvu == 0) {
            if (fulln)                               // stream next chunk while this one is used
                tdm_issue(in + (size_t)row0 * C + nx * CHUNK,
                          (unsigned)(size_t)&tile[nx & 1][0], C);
            if (fullc) {                             // TDM in-order per wave: <=1 left == ci done
                if (fulln) __builtin_amdgcn_s_wait_tensorcnt((short)1);
                else       __builtin_amdgcn_s_wait_tensorcnt((short)0);
            }
        }
        if (!fullc)
#endif
        {   // -------- float4 staging path (tail chunk, or no-TDM fallback) --------
            const int n4 = cc >> 2;
            for (int i = tid; i < 16 * n4; i += 256) {
                const int r = i / n4, c4 = i - r * n4;
                const float4 v = *(const float4*)(in + (size_t)(row0 + r) * C + cbase + 4 * c4);
                *(float4*)(&buf[r * LSTR + 4 * c4]) = v;
            }
        }
        __syncthreads();

        // -------- per-chunk row max: wave w handles rows 2w, 2w+1 --------
        float m0 = -__builtin_inff(), m1 = -__builtin_inff();
        for (int ccol = lane; ccol < cc; ccol += 32) {
            m0 = fmaxf(m0, buf[(2 * wvu)     * LSTR + ccol]);
            m1 = fmaxf(m1, buf[(2 * wvu + 1) * LSTR + ccol]);
        }
        #pragma unroll
        for (int off = 16; off; off >>= 1) {
            m0 = fmaxf(m0, __shfl_xor(m0, off));
            m1 = fmaxf(m1, __shfl_xor(m1, off));
        }
        if (lane == 0) {
            int r = 2 * wvu;
            float old = s_rowmax[r], nm = fmaxf(old, m0);
            s_scale[r] = __expf(old - nm); s_rowmax[r] = nm;
            r = 2 * wvu + 1;
            old = s_rowmax[r]; nm = fmaxf(old, m1);
            s_scale[r] = __expf(old - nm); s_rowmax[r] = nm;
        }
        __syncthreads();

        // -------- rescale accumulator to the new running max --------
        #pragma unroll
        for (int v = 0; v < 8; ++v) acc[v] *= s_scale[v + half];

        // -------- exp + WMMA 16x16x4 f32 accumulation (B = ones), scalar loop bounds --------
        const float mx = s_rowmax[m16];
        const float* lptr = &buf[m16 * LSTR + kb];
        const int ng = cc >> 2;                      // 4-column groups in chunk
        v2f b1; b1[0] = 1.0f; b1[1] = 1.0f;
        int g = wvu;
        for (; g + 8 < ng; g += 16) {                // unroll x2: two ds_loads in flight
            const float2 x0 = *(const float2*)(lptr + 4 * g);
            const float2 x1 = *(const float2*)(lptr + 4 * (g + 8));
            v2f a0, a1;
            a0[0] = __expf(x0.x - mx); a0[1] = __expf(x0.y - mx);
            a1[0] = __expf(x1.x - mx); a1[1] = __expf(x1.y - mx);
            acc = __builtin_amdgcn_wmma_f32_16x16x4_f32(false, a0, false, b1,
                                                        (short)0, acc, false, false);
            acc = __builtin_amdgcn_wmma_f32_16x16x4_f32(false, a1, false, b1,
                                                        (short)0, acc, false, false);
        }
        if (g < ng) {
            const float2 x0 = *(const float2*)(lptr + 4 * g);
            v2f a0;
            a0[0] = __expf(x0.x - mx); a0[1] = __expf(x0.y - mx);
            acc = __builtin_amdgcn_wmma_f32_16x16x4_f32(false, a0, false, b1,
                                                        (short)0, acc, false, false);
        }
        __syncthreads();
    }

    // -------- extract row sums from C layout (lane 0 -> rows 0..7, lane 16 -> rows 8..15) --------
    if (lane == 0 || lane == 16) {
        #pragma unroll
        for (int v = 0; v < 8; ++v) atomicAdd(&s_rowsum[v + half], acc[v]);
    }
    __syncthreads();

    // -------- finalize 16 rows --------
    if (tid < 16) {
        const int   n  = row0 + tid;
        const float mx = s_rowmax[tid];
        const float sm = s_rowsum[tid];
        const int   t  = tgt[n];
        const float xt = in[(size_t)n * C + t];
        const float lp = xt - mx - __logf(sm);
        g_logpt[n] = lp;
        g_key[n]   = __float_as_uint(__expf(lp));
    }
}

// ---------------- level-1 histogram of key>>16 ----------------
__global__ void k_hist1(int N) {
    for (int i = blockIdx.x * blockDim.x + threadIdx.x; i < N; i += gridDim.x * blockDim.x)
        atomicAdd(&g_hist1[g_key[i] >> 16], 1);
}

// ---------------- scan hist1, locate the NT quantile buckets ----------------
__global__ __launch_bounds__(1024) void k_scan1(int N) {
    __shared__ int ls[1024];
    const int t = threadIdx.x;
    const int bsz = N / NB;
    int s = 0;
    for (int j = 0; j < 64; ++j) s += g_hist1[t * 64 + j];
    ls[t] = s;
    __syncthreads();
    if (t == 0) { int run = 0; for (int i = 0; i < 1024; ++i) { int v = ls[i]; ls[i] = run; run += v; } }
    __syncthreads();
    const int base = ls[t];
    for (int k = 0; k < NT; ++k) {
        const int R = (k + 1) * bsz;                 // rank of threshold element
        if (R >= base && R < base + s) {
            int cum = base;
            for (int j = 0; j < 64; ++j) {
                int h = g_hist1[t * 64 + j];
                if (R < cum + h) { g_selb[k] = (unsigned)(t * 64 + j); g_r1[k] = R - cum; break; }
                cum += h;
            }
        }
    }
}

// ---------------- level-2 histogram (low 16 bits) for selected buckets ----------------
__global__ void k_hist2(int N) {
    __shared__ unsigned sb[NT];
    if (threadIdx.x < NT) sb[threadIdx.x] = g_selb[threadIdx.x];
    __syncthreads();
    for (int i = blockIdx.x * blockDim.x + threadIdx.x; i < N; i += gridDim.x * blockDim.x) {
        const unsigned key = g_key[i];
        const unsigned hb = key >> 16, lb = key & 0xFFFFu;
        #pragma unroll
        for (int j = 0; j < NT; ++j)
            if (hb == sb[j]) atomicAdd(&g_hist2[j * HB + (int)lb], 1);
    }
}

// ---------------- scan hist2 per threshold, get full key + residual ----------------
__global__ __launch_bounds__(1024) void k_scan2() {
    __shared__ int ls[1024];
    const int k = blockIdx.x;
    const int t = threadIdx.x;
    const int* h = g_hist2 + k * HB;
    int s = 0;
    for (int j = 0; j < 64; ++j) s += h[t * 64 + j];
    ls[t] = s;
    __syncthreads();
    if (t == 0) { int run = 0; for (int i = 0; i < 1024; ++i) { int v = ls[i]; ls[i] = run; run += v; } }
    __syncthreads();
    const int base = ls[t];
    const int R = g_r1[k];
    if (R >= base && R < base + s) {
        int cum = base;
        for (int j = 0; j < 64; ++j) {
            int hv = h[t * 64 + j];
            if (R < cum + hv) {
                g_tkey[k] = (g_selb[k] << 16) | (unsigned)(t * 64 + j);
                g_r2[k] = R - cum;
                break;
            }
            cum += hv;
        }
    }
}

// ---------------- tie-break by original index (stable-sort semantics) ----------------
__global__ void k_tie(int N) {
    __shared__ int buf[TIE_CAP];
    __shared__ int cnt;
    const int k = blockIdx.x;
    if (threadIdx.x == 0) cnt = 0;
    __syncthreads();
    const unsigned tkey = g_tkey[k];
    for (int i = threadIdx.x; i < N; i += blockDim.x) {
        if (g_key[i] == tkey) {
            int p = atomicAdd(&cnt, 1);
            if (p < TIE_CAP) buf[p] = i;
        }
    }
    __syncthreads();
    const int m = min(cnt, TIE_CAP);
    const int r2 = g_r2[k];
    if (r2 >= m) {
        if (threadIdx.x == 0) g_tidx[k] = (m > 0) ? (unsigned)buf[0] : 0u;
    } else {
        for (int j = threadIdx.x; j < m; j += blockDim.x) {
            const int v = buf[j];
            int less = 0;
            for (int l = 0; l < m; ++l) less += (buf[l] < v);
            if (less == r2) g_tidx[k] = (unsigned)v;   // indices unique -> one writer
        }
    }
}

// ---------------- final elementwise loss + reduction ----------------
__global__ __launch_bounds__(256) void k_final(const int* __restrict__ tgt,
                                               const float* __restrict__ bin_ece,
                                               const float* __restrict__ ada_gaps,
                                               const float* __restrict__ cls_ece,
                                               const int* __restrict__ epoch,
                                               float* out, int N) {
    __shared__ float    sred[8];
    __shared__ unsigned s_tk[NT];
    __shared__ unsigned s_ti[NT];
    if (threadIdx.x < NT) { s_tk[threadIdx.x] = g_tkey[threadIdx.x]; s_ti[threadIdx.x] = g_tidx[threadIdx.x]; }
    __syncthreads();

    const int i = blockIdx.x * blockDim.x + threadIdx.x;
    float acc = 0.0f;
    if (i < N) {
        const float lp  = g_logpt[i];
        const unsigned key = g_key[i];
        const float prob = __uint_as_float(key);
        // fixed equal-width binning (searchsorted right - 1, clipped)
        int fidx = 0;
        #pragma unroll
        for (int j = 1; j < NB; ++j) fidx += (prob >= (float)j * (1.0f / (float)NB));
        fidx += (prob >= 1.0f);
        if (fidx > NB - 1) fidx = NB - 1;
        const float ece = bin_ece[fidx];
        // adaptive equal-mass bin: count thresholds with composite key <= ours
        int aidx = 0;
        #pragma unroll
        for (int k = 0; k < NT; ++k) {
            const unsigned tk = s_tk[k];
            aidx += (int)((key > tk) | ((key == tk) & ((unsigned)i >= s_ti[k])));
        }
        const float ada = ada_gaps[aidx];
        const float cls = cls_ece[tgt[i]];
        const float w   = (fabsf(ece) + fabsf(ada) + 2.0f * fabsf(cls)) * (1.0f / 3.0f);
        const float lam = 1.0f - (float)epoch[0] / 350.0f;
        const float wt  = lam + (1.0f - lam) * w;
        acc = -wt * lp;
    }
    #pragma unroll
    for (int off = 16; off; off >>= 1) acc += __shfl_xor(acc, off);
    const int lane = threadIdx.x & 31, wv = threadIdx.x >> 5;
    if (lane == 0) sred[wv] = acc;
    __syncthreads();
    if (wv == 0) {
        float v = (lane < 8) ? sred[lane] : 0.0f;
        #pragma unroll
        for (int off = 4; off; off >>= 1) v += __shfl_xor(v, off);
        if (lane == 0) atomicAdd(out, v);
    }
}

// ---------------- host launcher ----------------
extern "C" void kernel_launch(void* const* d_in, const int* in_sizes, int n_in,
                              void* d_out, int out_size, void* d_ws, size_t ws_size,
                              hipStream_t stream) {
    const float* in     = (const float*)d_in[0];
    const int*   tgt    = (const int*)  d_in[1];
    const int*   epoch  = (const int*)  d_in[2];
    const float* binece = (const float*)d_in[3];
    const float* ada    = (const float*)d_in[4];
    const float* cls    = (const float*)d_in[5];
    float* out = (float*)d_out;

    const int N = in_sizes[1];                   // 131072
    const int C = in_sizes[5];                   // 1000

    k_zero     <<<(NT * HB) / 256, 256, 0, stream>>>(out);
    k_logsumexp<<<N / 16,          256, 0, stream>>>(in, tgt, N, C);
    k_hist1    <<<512,             256, 0, stream>>>(N);
    k_scan1    <<<1,              1024, 0, stream>>>(N);
    k_hist2    <<<512,             256, 0, stream>>>(N);
    k_scan2    <<<NT,             1024, 0, stream>>>();
    k_tie      <<<NT,              256, 0, stream>>>(N);
    k_final    <<<(N + 255) / 256, 256, 0, stream>>>(tgt, binece, ada, cls, epoch, out, N);
}